// IouDots_15187004359196
// MI455X (gfx1250) — compile-verified
//
#include <hip/hip_runtime.h>

#define IOU_EPS 1e-7f

typedef __attribute__((ext_vector_type(2))) float v2f;
typedef __attribute__((ext_vector_type(4))) float v4f;
typedef __attribute__((ext_vector_type(8))) float v8f;

// Full 32-lane (wave32) f32 sum using two V_WMMA_F32_16X16X4_F32.
// Requires EXEC == all ones (call only from uniform control flow).
__device__ __forceinline__ float wave_reduce_sum(float x) {
#if __has_builtin(__builtin_amdgcn_wmma_f32_16x16x4_f32)
    v2f a;    a[0] = x;    a[1] = 0.0f;
    v2f ones; ones[0] = 1.0f; ones[1] = 1.0f;
    v8f c = {};
    // D[m][n] = x[lane m] + x[lane m+16]   (A-side data, B = ones)
    v8f d = __builtin_amdgcn_wmma_f32_16x16x4_f32(
        false, a, false, ones, (short)0, c, false, false);
    // per-lane: rows 0-7 (lanes 0-15) or rows 8-15 (lanes 16-31)
    float t = d[0] + d[1] + d[2] + d[3] + d[4] + d[5] + d[6] + d[7];
    v2f a2; a2[0] = t; a2[1] = 0.0f;
    v8f d2 = __builtin_amdgcn_wmma_f32_16x16x4_f32(
        false, a2, false, ones, (short)0, c, false, false);
    return d2[0];   // every lane now holds the full 32-lane sum
#else
    for (int off = 16; off > 0; off >>= 1) x += __shfl_xor(x, off, 32);
    return x;
#endif
}

// Branchless per-box contribution: adds to the three running sums.
__device__ __forceinline__ void iou_box(v4f p, v4f g,
                                        float& mse_s, float& iou_s, float& inc_c) {
    // gt corners (cx,cy,w,h -> xyxy)
    float x_min_t = g[0] - g[2] * 0.5f;
    float x_max_t = g[0] + g[2] * 0.5f;
    float y_min_t = g[1] - g[3] * 0.5f;
    float y_max_t = g[1] + g[3] * 0.5f;
    // pred corners, clipped to [0,1]
    float x_min_p = fmaxf(p[0] - p[2] * 0.5f, 0.0f);
    float x_max_p = fminf(p[0] + p[2] * 0.5f, 1.0f);
    float y_min_p = fmaxf(p[1] - p[3] * 0.5f, 0.0f);
    float y_max_p = fminf(p[1] + p[3] * 0.5f, 1.0f);
    // overlap box
    float o0 = fmaxf(x_min_t, x_min_p);
    float o1 = fmaxf(y_min_t, y_min_p);
    float o2 = fminf(x_max_t, x_max_p);
    float o3 = fminf(y_max_t, y_max_p);

    float w_inc = ((o2 < o0) || (o3 < o1)) ? 1.0f : 0.0f;   // v_cndmask, no branch

    // MSE term (weighted by w_inc)
    float dx = p[0] - g[0], dy = p[1] - g[1], dz = p[2] - g[2], dw = p[3] - g[3];
    float sq = dx * dx + dy * dy + dz * dz + dw * dw;

    // IoU term (weighted by 1 - w_inc); denominator strictly positive
    float inter  = (o2 - o0) * (o3 - o1);
    float area_p = p[2] * p[3];
    float area_g = g[2] * g[3];
    float iou    = inter / (area_p + area_g - inter + IOU_EPS);

    mse_s += w_inc * sq;
    iou_s += (1.0f - w_inc) * iou;
    inc_c += w_inc;
}

__global__ void iou_dots_init(float* __restrict__ acc) {
    if (threadIdx.x < 4) acc[threadIdx.x] = 0.0f;
}

__global__ void iou_dots_main(const v4f* __restrict__ pr,
                              const v4f* __restrict__ gt,
                              float* __restrict__ acc, int n) {
    float mse_s = 0.0f, iou_s = 0.0f, inc_c = 0.0f, seen = 0.0f;

    const int stride = (int)(gridDim.x * blockDim.x);
    int i = (int)(blockIdx.x * blockDim.x + threadIdx.x);

    // 2x unrolled grid-stride: four independent b128 NT loads in flight
    for (; i + stride < n; i += 2 * stride) {
        v4f p0 = __builtin_nontemporal_load(pr + i);
        v4f g0 = __builtin_nontemporal_load(gt + i);
        v4f p1 = __builtin_nontemporal_load(pr + i + stride);
        v4f g1 = __builtin_nontemporal_load(gt + i + stride);
        iou_box(p0, g0, mse_s, iou_s, inc_c);
        iou_box(p1, g1, mse_s, iou_s, inc_c);
        seen += 2.0f;
    }
    if (i < n) {
        v4f p0 = __builtin_nontemporal_load(pr + i);
        v4f g0 = __builtin_nontemporal_load(gt + i);
        iou_box(p0, g0, mse_s, iou_s, inc_c);
        seen += 1.0f;
    }
    float corr_c = seen - inc_c;

    // Uniform point: EXEC all ones -> WMMA wave reduction legal.
    mse_s  = wave_reduce_sum(mse_s);
    iou_s  = wave_reduce_sum(iou_s);
    inc_c  = wave_reduce_sum(inc_c);
    corr_c = wave_reduce_sum(corr_c);

    __shared__ float red[4];
    if (threadIdx.x < 4) red[threadIdx.x] = 0.0f;
    __syncthreads();
    if ((threadIdx.x & 31u) == 0u) {          // one lane per wave32
        atomicAdd(&red[0], mse_s);            // ds_add_f32
        atomicAdd(&red[1], iou_s);
        atomicAdd(&red[2], inc_c);
        atomicAdd(&red[3], corr_c);
    }
    __syncthreads();
    if (threadIdx.x == 0) {                   // one atomic set per block
        atomicAdd(&acc[0], red[0]);           // global_atomic_add_f32
        atomicAdd(&acc[1], red[1]);
        atomicAdd(&acc[2], red[2]);
        atomicAdd(&acc[3], red[3]);
    }
}

__global__ void iou_dots_finalize(const float* __restrict__ acc,
                                  float* __restrict__ out) {
    float mse_sum = acc[0];
    float iou_sum = acc[1];
    float n_inc   = acc[2];   // integer-valued, exact in f32 (<= 2^23)
    float n_corr  = acc[3];
    float mse_mean = mse_sum / fmaxf(n_inc * 4.0f, 1.0f);
    float iou_mean = iou_sum / fmaxf(n_corr, 1.0f);
    float res_full = iou_mean + ((n_inc > 0.0f) ? -mse_mean : 0.0f);
    out[0] = (n_corr > 0.0f) ? res_full : -mse_mean;
}

extern "C" void kernel_launch(void* const* d_in, const int* in_sizes, int n_in,
                              void* d_out, int out_size, void* d_ws, size_t ws_size,
                              hipStream_t stream) {
    (void)n_in; (void)out_size; (void)ws_size;
    const v4f* pr = (const v4f*)d_in[0];
    const v4f* gt = (const v4f*)d_in[1];
    float* acc = (float*)d_ws;     // 4 floats: mse_sum, iou_sum, n_inc, n_corr
    float* out = (float*)d_out;

    const int n = in_sizes[0] / 4; // boxes (inputs are N x 4 f32)

    iou_dots_init<<<1, 4, 0, stream>>>(acc);

    const int threads = 256;       // 8 wave32 per block
    int blocks = (n + threads - 1) / threads;
    if (blocks > 2048) blocks = 2048;   // ~16K waves: enough to saturate HBM
    if (blocks < 1) blocks = 1;
    iou_dots_main<<<blocks, threads, 0, stream>>>(pr, gt, acc, n);

    iou_dots_finalize<<<1, 1, 0, stream>>>(acc, out);
}